// LIFSpike_66443144069281
// MI455X (gfx1250) — compile-verified
//
#include <hip/hip_runtime.h>

// LIF spiking-neuron forward scan, memory-bound (256 MB traffic, ~11us @ 23.3 TB/s).
// CDNA5 path: coalesced global_load_async_to_lds_b128 staging (double buffered,
// s_wait_asynccnt pipelining), padded-LDS ds_load_b128 reads, b128 stores.

#define T_STEPS    32
#define ROW_BYTES  128                 // T_STEPS * sizeof(float)
#define ROW_PITCH  144                 // +16B pad -> conflict-free LDS banks
#define WAVE_ROWS  32                  // one row per lane (wave32)
#define NWAVES     8
#define BLOCK_ROWS 256                 // NWAVES * WAVE_ROWS
#define WAVE_LDS   (WAVE_ROWS * ROW_PITCH)   // 4608 B per wave tile
#define BUF_LDS    (NWAVES * WAVE_LDS)       // 36864 B per buffer

__device__ __forceinline__ void async_ld_b128(unsigned ldsoff, unsigned long long gaddr) {
    // gfx1250 async DMA: LDS byte address in VGPR, 64-bit global address in VGPR pair.
    asm volatile("global_load_async_to_lds_b128 %0, %1, off"
                 :: "v"(ldsoff), "v"(gaddr) : "memory");
}

__device__ __forceinline__ void lif4(float& u, float& o, const float4 xv, float4& ov,
                                     const float tf, const float vth) {
    // u = tau*(u - Vth*o) + x ; o = (u - Vth > 0)  (exact reference FP ordering)
    u = fmaf(tf, fmaf(-vth, o, u), xv.x); o = ((u - vth) > 0.0f) ? 1.0f : 0.0f; ov.x = o;
    u = fmaf(tf, fmaf(-vth, o, u), xv.y); o = ((u - vth) > 0.0f) ? 1.0f : 0.0f; ov.y = o;
    u = fmaf(tf, fmaf(-vth, o, u), xv.z); o = ((u - vth) > 0.0f) ? 1.0f : 0.0f; ov.z = o;
    u = fmaf(tf, fmaf(-vth, o, u), xv.w); o = ((u - vth) > 0.0f) ? 1.0f : 0.0f; ov.w = o;
}

__global__ __launch_bounds__(BLOCK_ROWS)
void lif_scan_kernel(const float* __restrict__ x,
                     const float* __restrict__ Vth,
                     const float* __restrict__ tau,
                     float* __restrict__ out,
                     unsigned nrows, unsigned F, unsigned ntiles)
{
    __shared__ __align__(16) unsigned char smem[2 * BUF_LDS];   // 73728 B

    const unsigned tid  = threadIdx.x;
    const unsigned lane = tid & 31u;
    const unsigned wave = tid >> 5;

    // Low 32 bits of a generic pointer into __shared__ == LDS byte offset.
    const unsigned smem_base = (unsigned)(unsigned long long)(void*)smem;
    const unsigned wave_lds  = smem_base + wave * WAVE_LDS;
    // lane's slot inside one 512B coalesced chunk: row = lane/8, 16B col = lane%8
    const unsigned lds_lane  = (lane >> 3) * ROW_PITCH + (lane & 7u) * 16u;
    const unsigned long long xb = (unsigned long long)(const void*)x;

    unsigned tile = blockIdx.x;
    const unsigned tstride = gridDim.x;

    // ---- prologue: async-copy first tile into buffer 0 ----
    if (tile < ntiles) {
        unsigned long long g = xb
            + (unsigned long long)(tile * BLOCK_ROWS + wave * WAVE_ROWS) * ROW_BYTES
            + lane * 16u;
        unsigned l = wave_lds + lds_lane;
        #pragma unroll
        for (int c = 0; c < 8; ++c) {
            async_ld_b128(l, g);
            l += 4u * ROW_PITCH;      // 4 rows per 512B chunk
            g += 512u;
        }
    }

    unsigned buf = 0;
    for (; tile < ntiles; tile += tstride) {
        const unsigned next = tile + tstride;
        if (next < ntiles) {
            // our ds reads of the buffer we are about to overwrite are long done,
            // but make the ordering explicit before the DMA engine touches LDS
            asm volatile("s_wait_dscnt 0x0" ::: "memory");
            const unsigned nbuf = buf ^ 1u;
            unsigned long long g = xb
                + (unsigned long long)(next * BLOCK_ROWS + wave * WAVE_ROWS) * ROW_BYTES
                + lane * 16u;
            unsigned l = wave_lds + nbuf * BUF_LDS + lds_lane;
            #pragma unroll
            for (int c = 0; c < 8; ++c) {
                async_ld_b128(l, g);
                l += 4u * ROW_PITCH;
                g += 512u;
            }
            // 16 in flight; async loads complete in order -> wait for older 8
            asm volatile("s_wait_asynccnt 0x8" ::: "memory");
        } else {
            asm volatile("s_wait_asynccnt 0x0" ::: "memory");
        }

        // ---- compute: one row per lane, sequential T=32 LIF scan ----
        const unsigned row = tile * BLOCK_ROWS + wave * WAVE_ROWS + lane;
        const unsigned f   = row % F;
        const float tf  = tau[f];
        const float vth = Vth[f];
        const float4* lrow = (const float4*)(smem + buf * BUF_LDS
                                             + wave * WAVE_LDS + lane * ROW_PITCH);
        float4* __restrict__ orow = (float4*)(out + (unsigned long long)row * T_STEPS);
        float u = 0.0f, o = 0.0f;
        #pragma unroll
        for (int c = 0; c < 8; ++c) {
            const float4 xv = lrow[c];     // ds_load_b128, padded pitch -> no conflicts
            float4 ov;
            lif4(u, o, xv, ov, tf, vth);
            orow[c] = ov;                  // global_store_b128
        }
        buf ^= 1u;
    }

    // ---- tail: rows past the last full 256-row tile (empty for 512x2048x32) ----
    for (unsigned long long row = (unsigned long long)ntiles * BLOCK_ROWS
                                  + (unsigned long long)blockIdx.x * BLOCK_ROWS + tid;
         row < nrows;
         row += (unsigned long long)tstride * BLOCK_ROWS) {
        const unsigned f = (unsigned)(row % F);
        const float tf  = tau[f];
        const float vth = Vth[f];
        const float4* xr = (const float4*)(x + row * T_STEPS);
        float4* orow = (float4*)(out + row * T_STEPS);
        float u = 0.0f, o = 0.0f;
        #pragma unroll
        for (int c = 0; c < 8; ++c) {
            const float4 xv = xr[c];
            float4 ov;
            lif4(u, o, xv, ov, tf, vth);
            orow[c] = ov;
        }
    }
}

extern "C" void kernel_launch(void* const* d_in, const int* in_sizes, int n_in,
                              void* d_out, int out_size, void* d_ws, size_t ws_size,
                              hipStream_t stream) {
    const float* x   = (const float*)d_in[0];   // [B, F, T] f32
    const float* Vth = (const float*)d_in[1];   // [F] f32
    const float* tau = (const float*)d_in[2];   // [F] f32
    float* out = (float*)d_out;                 // [B, F, T] f32

    const long long total = (long long)in_sizes[0];
    const unsigned F      = (unsigned)in_sizes[1];
    const unsigned nrows  = (unsigned)(total / T_STEPS);      // B*F = 1,048,576
    const unsigned ntiles = nrows / BLOCK_ROWS;               // 4096

    // Persistent grid: 512 blocks x 8 waves ~= fills the chip; each block
    // pipelines ~8 tiles through the double-buffered async staging.
    unsigned grid = 512;
    if (ntiles > 0 && ntiles < grid) grid = ntiles;
    if (grid == 0) grid = 1;

    lif_scan_kernel<<<grid, BLOCK_ROWS, 0, stream>>>(x, Vth, tau, out,
                                                     nrows, F, ntiles);
}